// DeformRoIPooling_14516989461142
// MI455X (gfx1250) — compile-verified
//
#include <hip/hip_runtime.h>

typedef float v2f __attribute__((ext_vector_type(2)));
typedef float v8f __attribute__((ext_vector_type(8)));

#define BATCH 4
#define CH 256
#define HH 64
#define WW 64
#define NROI 256
#define SPATIAL_SCALE 0.125f
#define TRANS_STD 0.1f
#define BATCH_STRIDE (CH * HH * WW)   // 1,048,576 elements for both NCHW and NHWC

// ---------------------------------------------------------------------------
// Pass 1: NCHW -> NHWC transpose through LDS (makes channel gathers contiguous)
// ---------------------------------------------------------------------------
__global__ __launch_bounds__(256) void nchw_to_nhwc(const float* __restrict__ src,
                                                    float* __restrict__ dst) {
    __shared__ float tile[32][33];
    const int tx = threadIdx.x & 31;
    const int ty = threadIdx.x >> 5;
    const int hw0 = blockIdx.x * 32;
    const int c0  = blockIdx.y * 32;
    const int b   = blockIdx.z;
    const float* s = src + (size_t)b * BATCH_STRIDE;
    float*       d = dst + (size_t)b * BATCH_STRIDE;
#pragma unroll
    for (int i = 0; i < 4; ++i) {
        const int c = c0 + ty + i * 8;
        tile[ty + i * 8][tx] = s[(size_t)c * (HH * WW) + hw0 + tx];
    }
    __syncthreads();
#pragma unroll
    for (int i = 0; i < 4; ++i) {
        const int hw = hw0 + ty + i * 8;
        d[(size_t)hw * CH + c0 + tx] = tile[tx][ty + i * 8];
    }
}

// ---------------------------------------------------------------------------
// Pass 2: wave-per-bin deformable ROI pooling.
//   Masked 16-sample reduction done with chained V_WMMA_F32_16X16X4_F32:
//   D += A(16x4: broadcast mask rows) * B(4x16: sample values per channel).
//   Lanes 0-15 own K={0,1} (samples 4j+0,4j+1), lanes 16-31 own K={2,3}
//   (samples 4j+2,4j+3) per the ISA A/B VGPR layouts -> no redundant gathers.
// ---------------------------------------------------------------------------
__global__ __launch_bounds__(256) void deform_roi_pool_wmma(
    const float* __restrict__ src, const float* __restrict__ rois,
    const float* __restrict__ offs, float* __restrict__ out,
    int pix_stride, int ch_stride) {
    __shared__ float sout[CH * 49];

    const int n    = blockIdx.x;
    const int lane = threadIdx.x & 31;
    const int wave = threadIdx.x >> 5;
    const int half = lane >> 4;   // which lane half -> which K sub-pair
    const int cl   = lane & 15;   // channel column within the 16-wide group

    // ROI geometry (wave-uniform)
    const float* r = rois + n * 5;
    const int   bidx   = (int)r[0];
    const float roi_sw = rintf(r[1]) * SPATIAL_SCALE - 0.5f;
    const float roi_sh = rintf(r[2]) * SPATIAL_SCALE - 0.5f;
    const float roi_ew = (rintf(r[3]) + 1.0f) * SPATIAL_SCALE - 0.5f;
    const float roi_eh = (rintf(r[4]) + 1.0f) * SPATIAL_SCALE - 0.5f;
    const float roi_w  = fmaxf(roi_ew - roi_sw, 0.1f);
    const float roi_h  = fmaxf(roi_eh - roi_sh, 0.1f);
    const float bin_w  = roi_w * (1.0f / 7.0f);
    const float bin_h  = roi_h * (1.0f / 7.0f);
    const float sub_w  = bin_w * 0.25f;
    const float sub_h  = bin_h * 0.25f;
    const float* src_b = src + (size_t)bidx * BATCH_STRIDE;

    for (int bin = wave; bin < 49; bin += 8) {
        const int ph = bin / 7, pw = bin % 7;
        // P == PART == 7 -> part_h == ph, part_w == pw
        const float tx = offs[n * 98 + ph * 7 + pw] * TRANS_STD;
        const float ty = offs[n * 98 + 49 + ph * 7 + pw] * TRANS_STD;
        const float wstart = pw * bin_w + roi_sw + tx * roi_w;
        const float hstart = ph * bin_h + roi_sh + ty * roi_h;

        int   base_[8], rs_[8], cs_[8];
        float w00_[8], w01_[8], w10_[8], w11_[8], msk_[8];
        int count = 0;
#pragma unroll
        for (int s = 0; s < 16; ++s) {
            const int sy = s >> 2, sx = s & 3;
            const float w = wstart + sx * sub_w;
            const float h = hstart + sy * sub_h;
            const bool valid = (w > -0.5f) && (w < WW - 0.5f) &&
                               (h > -0.5f) && (h < HH - 0.5f);
            count += valid ? 1 : 0;
            if (((s >> 1) & 1) == half) {          // this lane-half owns sample s
                const int li = (s >> 2) * 2 + (s & 1);
                const float wc = fminf(fmaxf(w, 0.0f), (float)(WW - 1));
                const float hc = fminf(fmaxf(h, 0.0f), (float)(HH - 1));
                const int x0 = (int)floorf(wc), y0 = (int)floorf(hc);
                const int x1 = min(x0 + 1, WW - 1), y1 = min(y0 + 1, HH - 1);
                const float dx = wc - (float)x0, dy = hc - (float)y0;
                base_[li] = (y0 * WW + x0) * pix_stride;
                rs_[li]   = (y1 - y0) * WW * pix_stride;
                cs_[li]   = (x1 - x0) * pix_stride;
                w00_[li]  = (1.0f - dx) * (1.0f - dy);
                w10_[li]  = (1.0f - dx) * dy;        // (y1, x0)
                w01_[li]  = dx * (1.0f - dy);        // (y0, x1)
                w11_[li]  = dx * dy;                 // (y1, x1)
                msk_[li]  = valid ? 1.0f : 0.0f;
            }
        }
        const float scl = (count > 0) ? (1.0f / (float)count) : 0.0f;

        for (int cg = 0; cg < 16; ++cg) {
            const int c    = cg * 16 + cl;
            const int coff = c * ch_stride;
            v8f acc = {};
#pragma unroll
            for (int j = 0; j < 4; ++j) {
                v2f a, b;
                a.x = msk_[2 * j];
                a.y = msk_[2 * j + 1];
#pragma unroll
                for (int t = 0; t < 2; ++t) {
                    const int li = 2 * j + t;
                    const float* p = src_b + base_[li] + coff;
                    const float v = w00_[li] * p[0]
                                  + w01_[li] * p[cs_[li]]
                                  + w10_[li] * p[rs_[li]]
                                  + w11_[li] * p[rs_[li] + cs_[li]];
                    if (t == 0) b.x = v; else b.y = v;
                }
                // D = A*B + C : masked partial sum over samples 4j..4j+3
                acc = __builtin_amdgcn_wmma_f32_16x16x4_f32(
                    false, a, false, b, (short)0, acc, false, false);
            }
            // D[0, c] (elem 0, lanes 0-15) = sum_s m[s] * V[s, c]
            if (lane < 16) sout[c * 49 + bin] = acc[0] * scl;
        }
    }
    __syncthreads();
    // Coalesced contiguous store of this ROI's (C,7,7) slab
    float* on = out + (size_t)n * (CH * 49);
    for (int i = threadIdx.x; i < CH * 49; i += 256) on[i] = sout[i];
}

// ---------------------------------------------------------------------------
extern "C" void kernel_launch(void* const* d_in, const int* in_sizes, int n_in,
                              void* d_out, int out_size, void* d_ws, size_t ws_size,
                              hipStream_t stream) {
    (void)in_sizes; (void)n_in; (void)out_size;
    const float* data = (const float*)d_in[0];
    const float* rois = (const float*)d_in[1];
    const float* offs = (const float*)d_in[2];
    float* out = (float*)d_out;

    const size_t need = (size_t)BATCH * BATCH_STRIDE * sizeof(float);  // 16 MB
    const bool use_nhwc = (d_ws != nullptr) && (ws_size >= need);

    const float* src = data;
    int pix_stride = 1, ch_stride = HH * WW;   // NCHW fallback addressing
    if (use_nhwc) {
        dim3 g(HH * WW / 32, CH / 32, BATCH);
        nchw_to_nhwc<<<g, 256, 0, stream>>>(data, (float*)d_ws);
        src = (const float*)d_ws;
        pix_stride = CH; ch_stride = 1;        // NHWC addressing
    }
    deform_roi_pool_wmma<<<NROI, 256, 0, stream>>>(src, rois, offs, out,
                                                   pix_stride, ch_stride);
}